// DRIGNNBCE_28475633172831
// MI455X (gfx1250) — compile-verified
//
#include <hip/hip_runtime.h>
#include <hip/hip_bf16.h>

// ---------------------------------------------------------------------------
// DRIGNN-BCE forward for MI455X (gfx1250, wave32).
//  Phase 1: LightGCN 3-layer propagation (L2-resident, f32 atomics)
//  Phase 2: trend MLP via V_WMMA_F32_16X16X4_F32 (exact f32 matrix path)
//  Phase 3: per-row scoring with wave32 reductions
// ---------------------------------------------------------------------------

typedef float v2f __attribute__((ext_vector_type(2)));
typedef float v8f __attribute__((ext_vector_type(8)));

#define D_EMB   64
#define T_TRAIN 128
#define T_MAX   256
#define H_MLP   32
#define NLAYERS 3

// ---------------------------- utility kernels ------------------------------

__global__ void zero_f32_kernel(float* __restrict__ p, size_t n) {
    size_t i = (size_t)blockIdx.x * blockDim.x + threadIdx.x;
    size_t stride = (size_t)gridDim.x * blockDim.x;
    for (; i < n; i += stride) p[i] = 0.0f;
}

__global__ void zero_one_kernel(float* __restrict__ p) { *p = 0.0f; }

// emb = concat(user_emb, item_emb); acc = emb
__global__ void concat_init_kernel(const float* __restrict__ ue,
                                   const float* __restrict__ ie,
                                   float* __restrict__ emb,
                                   float* __restrict__ acc,
                                   int NU, size_t nd) {
    size_t i = (size_t)blockIdx.x * blockDim.x + threadIdx.x;
    size_t stride = (size_t)gridDim.x * blockDim.x;
    size_t ubytes = (size_t)NU * D_EMB;
    for (; i < nd; i += stride) {
        float v = (i < ubytes) ? ue[i] : ie[i - ubytes];
        emb[i] = v;
        acc[i] = v;
    }
}

__global__ void degree_kernel(const int* __restrict__ dst,
                              float* __restrict__ deg, int E) {
    int i = blockIdx.x * blockDim.x + threadIdx.x;
    if (i < E) unsafeAtomicAdd(&deg[dst[i]], 1.0f);
}

__global__ void invdeg_kernel(float* __restrict__ deg, int N) {
    int i = blockIdx.x * blockDim.x + threadIdx.x;
    if (i < N) {
        float d = deg[i];
        deg[i] = (d > 0.0f) ? (1.0f / d) : 0.0f;   // in-place inverse degree
    }
}

// scatter-add of gathered source features: one thread = (edge, 4 channels)
__global__ void scatter_kernel(const int* __restrict__ src,
                               const int* __restrict__ dst,
                               const float* __restrict__ cur,
                               float* __restrict__ nxt, int E) {
    int idx = blockIdx.x * blockDim.x + threadIdx.x;     // E * 16 threads
    if (idx >= E * 16) return;
    int e = idx >> 4;
    int q = (idx & 15) << 2;                              // channel group of 4
    int s = src[e];
    int t = dst[e];
    const float4 v = *(const float4*)&cur[(size_t)s * D_EMB + q];
    float* base = &nxt[(size_t)t * D_EMB + q];
    unsafeAtomicAdd(base + 0, v.x);
    unsafeAtomicAdd(base + 1, v.y);
    unsafeAtomicAdd(base + 2, v.z);
    unsafeAtomicAdd(base + 3, v.w);
}

// nxt *= inv_deg[node]; acc += nxt   (nxt becomes next layer's cur)
__global__ void scale_acc_kernel(float* __restrict__ nxt,
                                 const float* __restrict__ invd,
                                 float* __restrict__ acc, size_t nd) {
    size_t i = (size_t)blockIdx.x * blockDim.x + threadIdx.x;
    size_t stride = (size_t)gridDim.x * blockDim.x;
    for (; i < nd; i += stride) {
        size_t node = i >> 6;                             // D = 64
        float v = nxt[i] * invd[node];
        nxt[i] = v;
        acc[i] += v;
    }
}

// ------------------------------- MLP (WMMA) --------------------------------
// M = relu(T @ W1 + b1) @ W2 + b2, T:[B,128], W1:[128,32], W2:[32,128].
// One wave per 16-row batch tile; f32 WMMA 16x16x4 per ISA §7.12.2 layouts:
//   A frag : lane<16 -> row=lane, K={k0,k0+1}; lane>=16 -> row=lane-16, K={k0+2,k0+3}
//   B frag : lane half selects K-pair, lane&15 = column
//   C/D    : VGPR v, lane L -> (m = v + 8*(L>>4), n = L&15)
__global__ __launch_bounds__(32) void mlp_wmma_kernel(
        const float* __restrict__ T,    // [B, 128]
        const float* __restrict__ W1,   // [128, 32]
        const float* __restrict__ b1,   // [32]
        const float* __restrict__ W2,   // [32, 128]
        const float* __restrict__ b2,   // [128]
        float* __restrict__ M) {        // [B, 128]
    __shared__ float Hs[16 * H_MLP];

    const int row0 = blockIdx.x * 16;
    const int lane = threadIdx.x;       // 0..31, wave32
    const int half = lane >> 4;         // 0 or 1
    const int l15  = lane & 15;

    // ---- GEMM1: H(16x32) = relu(T_tile(16x128) @ W1 + b1) ----
    for (int nt = 0; nt < H_MLP / 16; ++nt) {
        const int n0 = nt * 16;
        v8f c = {};
        for (int k0 = 0; k0 < T_TRAIN; k0 += 4) {
            v2f a = *(const v2f*)&T[(size_t)(row0 + l15) * T_TRAIN + k0 + 2 * half];
            v2f b;
            b.x = W1[(size_t)(k0 + 2 * half)     * H_MLP + n0 + l15];
            b.y = W1[(size_t)(k0 + 2 * half + 1) * H_MLP + n0 + l15];
            c = __builtin_amdgcn_wmma_f32_16x16x4_f32(
                    false, a, false, b, (short)0, c, false, false);
        }
        const float bias = b1[n0 + l15];
#pragma unroll
        for (int v = 0; v < 8; ++v) {
            float h = c[v] + bias;
            Hs[(v + 8 * half) * H_MLP + n0 + l15] = (h > 0.0f) ? h : 0.0f;
        }
    }
    __syncthreads();

    // ---- GEMM2: M(16x128) = H(16x32) @ W2 + b2 ----
    for (int nt = 0; nt < T_TRAIN / 16; ++nt) {
        const int n0 = nt * 16;
        v8f c = {};
        for (int k0 = 0; k0 < H_MLP; k0 += 4) {
            v2f a = *(const v2f*)&Hs[l15 * H_MLP + k0 + 2 * half];
            v2f b;
            b.x = W2[(size_t)(k0 + 2 * half)     * T_TRAIN + n0 + l15];
            b.y = W2[(size_t)(k0 + 2 * half + 1) * T_TRAIN + n0 + l15];
            c = __builtin_amdgcn_wmma_f32_16x16x4_f32(
                    false, a, false, b, (short)0, c, false, false);
        }
        const float bias = b2[n0 + l15];
#pragma unroll
        for (int v = 0; v < 8; ++v) {
            M[(size_t)(row0 + v + 8 * half) * T_TRAIN + n0 + l15] = c[v] + bias;
        }
    }
}

// ------------------------------ batch scoring ------------------------------
// One wave32 per batch row: embedding dot (final = acc/4 folded as 1/16),
// 256-dim time dots, reg-loss partials; wave shuffle reductions.
__global__ __launch_bounds__(256) void batch_kernel(
        const int* __restrict__ uidx, const int* __restrict__ iidx,
        const float* __restrict__ utr, const float* __restrict__ itr,  // [B,128]
        const float* __restrict__ ue0, const float* __restrict__ ie0,  // tables
        const float* __restrict__ utt, const float* __restrict__ itt,  // [*,256]
        const float* __restrict__ acc,                                 // [N,64]
        const float* __restrict__ Mu,  const float* __restrict__ Mi,   // [B,128]
        float* __restrict__ out, int B, int NU) {
    const int wave = (blockIdx.x * blockDim.x + threadIdx.x) >> 5;
    const int lane = threadIdx.x & 31;
    if (wave >= B) return;
    const int b = wave;
    const int u = uidx[b];
    const int v = iidx[b];

    const float* au = acc + (size_t)u * D_EMB;
    const float* ai = acc + (size_t)(NU + v) * D_EMB;
    const float* eu = ue0 + (size_t)u * D_EMB;
    const float* ei = ie0 + (size_t)v * D_EMB;

    float gm = 0.0f, reg = 0.0f;
#pragma unroll
    for (int d = lane; d < D_EMB; d += 32) {
        gm += au[d] * ai[d];
        float a = eu[d], c = ei[d];
        reg += a * a + c * c;
    }
    gm *= 0.0625f;   // (acc/4)·(acc/4)

    const float* tu = utt + (size_t)u * T_MAX;
    const float* ti = itt + (size_t)v * T_MAX;
    float utm = 0.0f, itm = 0.0f;
#pragma unroll
    for (int t = lane; t < T_MAX; t += 32) {
        float uv = (t < T_TRAIN) ? utr[(size_t)b * T_TRAIN + t]
                                 : Mu[(size_t)b * T_TRAIN + t - T_TRAIN];
        float iv = (t < T_TRAIN) ? itr[(size_t)b * T_TRAIN + t]
                                 : Mi[(size_t)b * T_TRAIN + t - T_TRAIN];
        float a = tu[t], c = ti[t];
        utm += a * uv;
        itm += c * iv;
        reg += a * a + c * c;
    }
    for (int off = 16; off > 0; off >>= 1) {
        gm  += __shfl_down(gm,  off, 32);
        utm += __shfl_down(utm, off, 32);
        itm += __shfl_down(itm, off, 32);
        reg += __shfl_down(reg, off, 32);
    }
    if (lane == 0) {
        out[b]     = 1.0f / (1.0f + __expf(-gm));
        out[B + b] = 1.0f / (1.0f + __expf(-(utm + itm)));
        unsafeAtomicAdd(&out[2 * B], reg * 0.5f / (float)B);
    }
}

// ------------------------------- launcher ----------------------------------

extern "C" void kernel_launch(void* const* d_in, const int* in_sizes, int n_in,
                              void* d_out, int out_size, void* d_ws, size_t ws_size,
                              hipStream_t stream) {
    const int*   uidx = (const int*)  d_in[0];
    const int*   iidx = (const int*)  d_in[1];
    // d_in[2] (time_diffs) unused by the reference forward
    const float* utr  = (const float*)d_in[3];
    const float* itr  = (const float*)d_in[4];
    const float* uet  = (const float*)d_in[5];
    const float* iet  = (const float*)d_in[6];
    const float* utt  = (const float*)d_in[7];
    const float* itt  = (const float*)d_in[8];
    const float* W1   = (const float*)d_in[9];
    const float* b1   = (const float*)d_in[10];
    const float* W2   = (const float*)d_in[11];
    const float* b2   = (const float*)d_in[12];
    const int*   esrc = (const int*)  d_in[13];
    const int*   edst = (const int*)  d_in[14];
    float* out = (float*)d_out;

    const int B  = in_sizes[0];
    const int NU = in_sizes[5] / D_EMB;
    const int NI = in_sizes[6] / D_EMB;
    const int N  = NU + NI;
    const int E  = in_sizes[13];
    const size_t nd = (size_t)N * D_EMB;

    // workspace layout (floats): bufA | bufB | acc | deg(->invdeg)
    float* ws   = (float*)d_ws;
    float* bufA = ws;
    float* bufB = ws + nd;
    float* acc  = ws + 2 * nd;
    float* deg  = ws + 3 * nd;
    // MLP outputs carved from bufA (dead after the last graph layer)
    float* Mu = bufA;
    float* Mi = bufA + (size_t)B * T_TRAIN;

    const int TPB = 256;
    const int gn  = (int)((nd + TPB - 1) / TPB);

    // Phase 1: graph propagation
    zero_f32_kernel<<<(N + TPB - 1) / TPB, TPB, 0, stream>>>(deg, (size_t)N);
    concat_init_kernel<<<gn, TPB, 0, stream>>>(uet, iet, bufA, acc, NU, nd);
    degree_kernel<<<(E + TPB - 1) / TPB, TPB, 0, stream>>>(edst, deg, E);
    invdeg_kernel<<<(N + TPB - 1) / TPB, TPB, 0, stream>>>(deg, N);

    float* cur = bufA;
    float* nxt = bufB;
    const int scat_blocks = (E * 16 + TPB - 1) / TPB;
    for (int layer = 0; layer < NLAYERS; ++layer) {
        zero_f32_kernel<<<gn, TPB, 0, stream>>>(nxt, nd);
        scatter_kernel<<<scat_blocks, TPB, 0, stream>>>(esrc, edst, cur, nxt, E);
        scale_acc_kernel<<<gn, TPB, 0, stream>>>(nxt, deg, acc, nd);
        float* t = cur; cur = nxt; nxt = t;
    }
    // NLAYERS = 3 -> final cur == bufB, bufA is dead: safe to reuse for Mu/Mi.

    // Phase 2: trend MLP via f32 WMMA (one wave per 16 batch rows)
    mlp_wmma_kernel<<<B / 16, 32, 0, stream>>>(utr, W1, b1, W2, b2, Mu);
    mlp_wmma_kernel<<<B / 16, 32, 0, stream>>>(itr, W1, b1, W2, b2, Mi);

    // Phase 3: scoring (+ zero the reg-loss accumulator first)
    zero_one_kernel<<<1, 1, 0, stream>>>(&out[2 * B]);
    batch_kernel<<<(B * 32 + TPB - 1) / TPB, TPB, 0, stream>>>(
        uidx, iidx, utr, itr, uet, iet, utt, itt, acc, Mu, Mi, out, B, NU);
}